// GCNwithResConn_2241972929166
// MI455X (gfx1250) — compile-verified
//
#include <hip/hip_runtime.h>
#include <hip/hip_bf16.h>

// ---------------------------------------------------------------------------
// GCN with residual connections for MI455X (gfx1250, wave32).
// Dense transforms: V_WMMA_F32_16X16X4_F32 (full fp32 on the matrix pipes).
// One wave computes a 32x64 output strip (8 accumulators): the 8 B-fragment
// loads per k-step are amortized over 8 WMMAs (vs 4 before), and each A
// fragment feeds 4 WMMAs. Message passing keeps t/agg L2-resident (12.8MB
// each, 192MB L2) and uses b128 gathers + f32 atomics for the segment-sum.
// ---------------------------------------------------------------------------

#define IN_C 128
#define HID 64
#define OUT_C 64
#define NUM_LAYERS 8
#define NUM_DEC 2

typedef float v2f __attribute__((ext_vector_type(2)));
typedef float v4f __attribute__((ext_vector_type(4)));
typedef float v8f __attribute__((ext_vector_type(8)));

#define WMMA_F32(a, b, acc) \
    __builtin_amdgcn_wmma_f32_16x16x4_f32(false, (a), false, (b), (short)0, (acc), false, false)

// ---------------------------------------------------------------------------
// WMMA f32 GEMM: C[rows x 64] = A[rows x K] @ B[K x 64] (+ bias)
// One wave computes rows m0..m0+31 (all 64 output columns). If fewer than 32
// rows remain (rows is always a multiple of 16 here), a uniform-branch tail
// computes a single 16-row strip. EXEC stays all-ones on every WMMA.
//
// Fragment layouts per CDNA5 ISA 7.12.2 (wave32):
//   A 16x4 f32 : lane L -> m = L%16, VGPR0/1 = A[m][2*(L/16)+0 / +1]
//   B 4x16 f32 : lane L -> n = L%16, VGPR0/1 = B[2*(L/16)+0 / +1][n]
//   C/D 16x16  : lane L -> n = L%16, VGPR r = C[r + 8*(L/16)][n]
// ---------------------------------------------------------------------------
__global__ void gcn_gemm_wmma_f32(const float* __restrict__ A,
                                  const float* __restrict__ B,
                                  const float* __restrict__ bias, // may be null
                                  float* __restrict__ C,
                                  int rows, int K)
{
    const int lane = threadIdx.x & 31;
    const int wave = (int)((blockIdx.x * blockDim.x + threadIdx.x) >> 5);
    const int m0 = wave * 32;
    if (m0 >= rows) return;

    const int half = lane >> 4;                 // 0: lanes 0-15, 1: lanes 16-31
    const int l = lane & 15;

    float bv0 = 0.0f, bv1 = 0.0f, bv2 = 0.0f, bv3 = 0.0f;
    if (bias) {
        bv0 = bias[l];
        bv1 = bias[16 + l];
        bv2 = bias[32 + l];
        bv3 = bias[48 + l];
    }

    if (rows - m0 >= 32) {
        // ---- 32-row strip: 8 accumulators, B fragments shared by 2 A frags.
        v8f acc0 = {}, acc1 = {}, acc2 = {}, acc3 = {};
        v8f acc4 = {}, acc5 = {}, acc6 = {}, acc7 = {};
        const float* arow0 = A + (size_t)(m0 + l) * (size_t)K;
        const float* arow1 = arow0 + (size_t)16 * K;

#pragma unroll 2
        for (int k = 0; k < K; k += 4) {
            v2f a0 = *(const v2f*)(arow0 + k + 2 * half);
            v2f a1 = *(const v2f*)(arow1 + k + 2 * half);
            const float* b0 = B + (size_t)(k + 2 * half) * HID + l; // row k+2h
            const float* b1 = b0 + HID;                              // row k+2h+1
            v2f b;
            b.x = b0[0];  b.y = b1[0];
            acc0 = WMMA_F32(a0, b, acc0);
            acc4 = WMMA_F32(a1, b, acc4);
            b.x = b0[16]; b.y = b1[16];
            acc1 = WMMA_F32(a0, b, acc1);
            acc5 = WMMA_F32(a1, b, acc5);
            b.x = b0[32]; b.y = b1[32];
            acc2 = WMMA_F32(a0, b, acc2);
            acc6 = WMMA_F32(a1, b, acc6);
            b.x = b0[48]; b.y = b1[48];
            acc3 = WMMA_F32(a0, b, acc3);
            acc7 = WMMA_F32(a1, b, acc7);
        }

        float* crow0 = C + (size_t)(m0 + 8 * half) * HID + l;
        float* crow1 = crow0 + (size_t)16 * HID;
#pragma unroll
        for (int r = 0; r < 8; ++r) {
            crow0[(size_t)r * HID + 0]  = acc0[r] + bv0;
            crow0[(size_t)r * HID + 16] = acc1[r] + bv1;
            crow0[(size_t)r * HID + 32] = acc2[r] + bv2;
            crow0[(size_t)r * HID + 48] = acc3[r] + bv3;
            crow1[(size_t)r * HID + 0]  = acc4[r] + bv0;
            crow1[(size_t)r * HID + 16] = acc5[r] + bv1;
            crow1[(size_t)r * HID + 32] = acc6[r] + bv2;
            crow1[(size_t)r * HID + 48] = acc7[r] + bv3;
        }
    } else {
        // ---- 16-row tail strip (rows % 32 == 16).
        v8f acc0 = {}, acc1 = {}, acc2 = {}, acc3 = {};
        const float* arow = A + (size_t)(m0 + l) * (size_t)K;

#pragma unroll 4
        for (int k = 0; k < K; k += 4) {
            v2f a = *(const v2f*)(arow + k + 2 * half);
            const float* b0 = B + (size_t)(k + 2 * half) * HID + l;
            const float* b1 = b0 + HID;
            v2f b;
            b.x = b0[0];  b.y = b1[0];
            acc0 = WMMA_F32(a, b, acc0);
            b.x = b0[16]; b.y = b1[16];
            acc1 = WMMA_F32(a, b, acc1);
            b.x = b0[32]; b.y = b1[32];
            acc2 = WMMA_F32(a, b, acc2);
            b.x = b0[48]; b.y = b1[48];
            acc3 = WMMA_F32(a, b, acc3);
        }

        float* crow = C + (size_t)(m0 + 8 * half) * HID + l;
#pragma unroll
        for (int r = 0; r < 8; ++r) {
            crow[(size_t)r * HID + 0]  = acc0[r] + bv0;
            crow[(size_t)r * HID + 16] = acc1[r] + bv1;
            crow[(size_t)r * HID + 32] = acc2[r] + bv2;
            crow[(size_t)r * HID + 48] = acc3[r] + bv3;
        }
    }
}

// ---------------------------------------------------------------------------
// Degree / normalization kernels (GCN symmetric norm with self-loops).
// ---------------------------------------------------------------------------
__global__ void k_init_deg(float* __restrict__ deg, int n)
{
    int i = blockIdx.x * blockDim.x + threadIdx.x;
    if (i < n) deg[i] = 1.0f;                   // self-loop contributes 1
}

__global__ void k_accum_deg(const int* __restrict__ dst, float* __restrict__ deg, int e)
{
    int i = blockIdx.x * blockDim.x + threadIdx.x;
    if (i < e) atomicAdd(&deg[dst[i]], 1.0f);
}

__global__ void k_dinv(float* __restrict__ deg, int n)
{
    int i = blockIdx.x * blockDim.x + threadIdx.x;
    if (i < n) {
        float d = deg[i];
        deg[i] = (d > 0.0f) ? rsqrtf(d) : 0.0f; // in-place deg -> dinv
    }
}

__global__ void k_edge_norm(const int* __restrict__ src, const int* __restrict__ dst,
                            const float* __restrict__ dinv, float* __restrict__ norm, int e)
{
    int i = blockIdx.x * blockDim.x + threadIdx.x;
    if (i < e) norm[i] = dinv[src[i]] * dinv[dst[i]];
}

// ---------------------------------------------------------------------------
// agg init, fused residual + bias + self-loop message (vectorized float4):
//   agg[n][f] = h[n][f] + b[f] + t[n][f] * dinv[n]^2
// ---------------------------------------------------------------------------
__global__ void k_agg_init(const float* __restrict__ h, const float* __restrict__ t,
                           const float* __restrict__ dinv, const float* __restrict__ bias,
                           float* __restrict__ agg, int n)
{
    long i = (long)blockIdx.x * blockDim.x + threadIdx.x;
    long total = (long)n * (HID / 4);
    if (i >= total) return;
    int node = (int)(i >> 4);
    int f4 = (int)(i & 15) * 4;
    float dv = dinv[node];
    float d2 = dv * dv;
    v4f hv = *(const v4f*)(h + (size_t)node * HID + f4);
    v4f tv = *(const v4f*)(t + (size_t)node * HID + f4);
    v4f bv = *(const v4f*)(bias + f4);
    v4f out = hv + bv + tv * d2;
    *(v4f*)(agg + (size_t)node * HID + f4) = out;
}

// ---------------------------------------------------------------------------
// Edge scatter: agg[dst][f] += t[src][f] * norm[e].
// One thread per (edge, 4-feature chunk). b128 gather from L2-resident t,
// 4 f32 atomics resolving at L2.
// ---------------------------------------------------------------------------
__global__ void k_edge_scatter(const int* __restrict__ src, const int* __restrict__ dst,
                               const float* __restrict__ norm, const float* __restrict__ t,
                               float* __restrict__ agg, int e)
{
    long i = (long)blockIdx.x * blockDim.x + threadIdx.x;
    long total = (long)e * (HID / 4);
    if (i >= total) return;
    int edge = (int)(i >> 4);
    int f4 = (int)(i & 15) * 4;
    int s = src[edge];
    int d = dst[edge];
    float w = norm[edge];
    v4f tv = *(const v4f*)(t + (size_t)s * HID + f4);
    float* dest = agg + (size_t)d * HID + f4;
    atomicAdd(dest + 0, tv.x * w);
    atomicAdd(dest + 1, tv.y * w);
    atomicAdd(dest + 2, tv.z * w);
    atomicAdd(dest + 3, tv.w * w);
}

// ---------------------------------------------------------------------------
// Host-side orchestration.
// ---------------------------------------------------------------------------
extern "C" void kernel_launch(void* const* d_in, const int* in_sizes, int n_in,
                              void* d_out, int out_size, void* d_ws, size_t ws_size,
                              hipStream_t stream)
{
    const float* x     = (const float*)d_in[0];  // [N, 128]
    const int*   ei    = (const int*)d_in[1];    // [2, E]
    const float* encW  = (const float*)d_in[2];  // [128, 64]
    const float* encb  = (const float*)d_in[3];  // [64]
    const float* gcnW  = (const float*)d_in[4];  // [L, 64, 64]
    const float* gcnb  = (const float*)d_in[5];  // [L, 64]
    const float* decW  = (const float*)d_in[6];  // [2, 64, 64]
    const float* decb  = (const float*)d_in[7];  // [2, 64]
    const float* finW  = (const float*)d_in[8];  // [64, 64]
    const float* finb  = (const float*)d_in[9];  // [64]

    const int N = in_sizes[0] / IN_C;            // 50000
    const int E = in_sizes[1] / 2;               // 800000
    const int* src = ei;                         // edge_index[0]
    const int* dst = ei + E;                     // edge_index[1]

    // Workspace partition (floats): hA | hB | t | dinv | norm  (~42 MB)
    float* f = (float*)d_ws;
    const size_t nodeFeats = (size_t)N * HID;
    float* hA   = f;                  f += nodeFeats;
    float* hB   = f;                  f += nodeFeats;
    float* t    = f;                  f += nodeFeats;
    float* dinv = f;                  f += N;
    float* norm = f;                  f += E;
    (void)ws_size;

    const int BLK = 256;
    auto blocks1 = [&](long work) { return (int)((work + BLK - 1) / BLK); };
    // GEMM grid: one wave per 32-row strip (computes all 64 output columns).
    const int stripsM = (N + 31) / 32;
    const long gemmThreads = (long)stripsM * 32;
    const int gemmBlocks = blocks1(gemmThreads);

    // --- normalization ---
    k_init_deg <<<blocks1(N), BLK, 0, stream>>>(dinv, N);
    k_accum_deg<<<blocks1(E), BLK, 0, stream>>>(dst, dinv, E);
    k_dinv     <<<blocks1(N), BLK, 0, stream>>>(dinv, N);
    k_edge_norm<<<blocks1(E), BLK, 0, stream>>>(src, dst, dinv, norm, E);

    // --- encoder: h = x @ enc_W + enc_b ---
    gcn_gemm_wmma_f32<<<gemmBlocks, BLK, 0, stream>>>(x, encW, encb, hA, N, IN_C);

    float* h   = hA;
    float* nxt = hB;

    // --- 8 GCN layers with residual ---
    for (int i = 0; i < NUM_LAYERS; ++i) {
        const float* W = gcnW + (size_t)i * HID * HID;
        const float* b = gcnb + (size_t)i * HID;
        gcn_gemm_wmma_f32<<<gemmBlocks, BLK, 0, stream>>>(h, W, nullptr, t, N, HID);
        k_agg_init<<<blocks1((long)N * (HID / 4)), BLK, 0, stream>>>(h, t, dinv, b, nxt, N);
        k_edge_scatter<<<blocks1((long)E * (HID / 4)), BLK, 0, stream>>>(src, dst, norm, t, nxt, E);
        float* tmp = h; h = nxt; nxt = tmp;
    }

    // --- 2 decoder layers ---
    for (int i = 0; i < NUM_DEC; ++i) {
        const float* W = decW + (size_t)i * HID * HID;
        const float* b = decb + (size_t)i * HID;
        gcn_gemm_wmma_f32<<<gemmBlocks, BLK, 0, stream>>>(h, W, b, nxt, N, HID);
        float* tmp = h; h = nxt; nxt = tmp;
    }

    // --- final projection straight into d_out ---
    gcn_gemm_wmma_f32<<<gemmBlocks, BLK, 0, stream>>>(h, finW, finb, (float*)d_out, N, HID);
    (void)out_size; (void)n_in;
}